// EmbeddingBagCollection_56959856279962
// MI455X (gfx1250) — compile-verified
//
#include <hip/hip_runtime.h>

// EmbeddingBagCollection: T=8 tables, B=4096 bags/table, L=50 rows/bag,
// V=100000 rows/table, D=128 f32/row. out[b, t*D + d] = sum_l tab[t, idx[t,b,l], d].
//
// Roofline: ~838 MB gathered reads -> ~36 us @ 23.3 TB/s; ~210 MFLOP of adds is
// noise. Pure HBM-bound gather+reduce.
//
// Design:
//  - one wave32 per (t,b) bag; lane owns a float4 slice (32*4 = 128 = D), so each
//    row gather is one fully coalesced global_load_b128 over a contiguous 512 B row.
//  - bag indices are wave-uniform: readfirstlane the wave id so the 50 index loads
//    become s_load (KMcnt), keeping the vector pipe dedicated to row data.
//  - CDNA5 async data mover: the first 8 rows are gather-prefetched into per-wave
//    LDS scratch with global_load_async_to_lds_b128 (ASYNCcnt), overlapping the
//    scalar setup and warming GL2. The staging buffer is write-only scratch; the
//    accumulate loop uses direct b128 gathers (no LDS round-trip on the hot path).
//  - output (write-once, never re-read) uses non-temporal b128 stores so the
//    192 MB L2 keeps table rows, which have ~2x reuse (B*L/V ~= 2).

typedef float f32x4 __attribute__((ext_vector_type(4)));

namespace {
constexpr int kT = 8;
constexpr int kB = 4096;
constexpr int kL = 50;
constexpr int kV = 100000;
constexpr int kD = 128;
constexpr int kWavesPerBlock = 8;              // 256 threads = 8 wave32 = 8 bags/block
constexpr int kBlockThreads  = 32 * kWavesPerBlock;
constexpr int kStageRows     = 8;              // async-staged rows per bag
constexpr int kRowBytes      = kD * 4;         // 512 B
}

__global__ __launch_bounds__(kBlockThreads) void embag_sum_kernel(
    const int* __restrict__ indices,   // [T, B, L]
    const float* __restrict__ tables,  // [T, V, D]
    float* __restrict__ out)           // [B, T*D]
{
  // Per-wave LDS staging scratch for the async gather-prefetch (write-only).
  __shared__ __align__(16) char stage[kWavesPerBlock * kStageRows * kRowBytes];

  // Wave id forced scalar so the per-bag index pointer is provably uniform.
  const int wave = __builtin_amdgcn_readfirstlane((int)(threadIdx.x >> 5));
  const int lane = (int)(threadIdx.x & 31);

  const int bag = (int)blockIdx.x * kWavesPerBlock + wave;  // 0 .. T*B-1
  const int t   = bag / kB;
  const int b   = bag - t * kB;

  const int*   idxp = indices + ((size_t)t * kB + (size_t)b) * kL;  // uniform -> s_load
  const float* tab  = tables + (size_t)t * kV * kD;
  const int    col  = lane * 4;  // this lane's 4 contiguous columns

  // --- CDNA5 async engine: DMA the first kStageRows gathered rows into LDS. ---
  // Each instruction moves one 512 B row (32 lanes x 16 B), memory -> LDS, with
  // no VGPR landing zone; completion tracked by ASYNCcnt. This warms GL2 for the
  // direct-load loop below while the scalar index stream is still being fetched.
  {
    const unsigned ldsBase = (unsigned)(uintptr_t)(&stage[0]) +
                             (unsigned)(wave * kStageRows * kRowBytes) +
                             (unsigned)(lane * 16);
#pragma unroll
    for (int l = 0; l < kStageRows; ++l) {
      const int row = idxp[l];
      const unsigned long long gaddr =
          (unsigned long long)(uintptr_t)(tab + (size_t)row * kD + col);
      const unsigned laddr = ldsBase + (unsigned)(l * kRowBytes);
      asm volatile("global_load_async_to_lds_b128 %0, %1, off"
                   :
                   : "v"(laddr), "v"(gaddr)
                   : "memory");
    }
  }

  // --- Hot loop: direct coalesced b128 gathers, ~10 loads in flight. ---
  f32x4 acc = (f32x4)0.0f;
#pragma unroll 10
  for (int l = 0; l < kL; ++l) {
    const int row = idxp[l];  // scalar load
    const f32x4 v =
        *reinterpret_cast<const f32x4*>(tab + (size_t)row * kD + col);
    acc += v;
  }

  // Retire the async prefetches before wave exit (cheap; long since complete).
  asm volatile("s_wait_asynccnt 0x0" ::: "memory");

  // out[b, t*D + col .. +3]: 16 B aligned, coalesced, non-temporal b128 store.
  float* o = out + (size_t)b * (kT * kD) + (size_t)t * kD + col;
  __builtin_nontemporal_store(acc, reinterpret_cast<f32x4*>(o));
}

extern "C" void kernel_launch(void* const* d_in, const int* in_sizes, int n_in,
                              void* d_out, int out_size, void* d_ws, size_t ws_size,
                              hipStream_t stream) {
  (void)in_sizes; (void)n_in; (void)out_size; (void)d_ws; (void)ws_size;

  const int*   indices = (const int*)d_in[0];   // [T, B, L] (harness delivers int32)
  const float* tables  = (const float*)d_in[1]; // [T, V, D] f32
  float*       out     = (float*)d_out;         // [B, T*D]  f32

  constexpr int nBags  = kT * kB;               // 32768 bags
  const dim3 grid(nBags / kWavesPerBlock);      // 4096 blocks
  const dim3 block(kBlockThreads);              // 256 threads (8 wave32)

  embag_sum_kernel<<<grid, block, 0, stream>>>(indices, tables, out);
}